// Attention_6073083756792
// MI455X (gfx1250) — compile-verified
//
#include <hip/hip_runtime.h>

// ---------------------------------------------------------------------------
// Linear-attention collapse: since the reference applies NO softmax (bug
// faithfully reproduced), out = Q^T (K V^T) with K V^T a 32x32 matrix per
// (batch, head).  All GEMMs run on V_WMMA_F32_16X16X4_F32 (fp32, wave32).
// ---------------------------------------------------------------------------

typedef __attribute__((ext_vector_type(2))) float v2f;
typedef __attribute__((ext_vector_type(8))) float v8f;

#define NB    2
#define C_IN  64
#define NHEAD 4
#define DIMH  32
#define NQKV  384
#define NPOS  4096
#define SCALE 0.17677669529663687f   // 32^-0.5

static __device__ __forceinline__ v8f wmma4(v2f a, v2f b, v8f c) {
  // D = A(16x4) * B(4x16) + C(16x16), fp32
  return __builtin_amdgcn_wmma_f32_16x16x4_f32(
      false, a, false, b, (short)0, c, false, false);
}

// ---------------------------------------------------------------------------
// Kernel 1: RMSNorm over channels + 1x1-conv QKV projection.
// Block = 256 threads (8 waves), handles 64 positions of one batch.
// qkv stored channel-major [b][384][4096] (Q rows 0..127 pre-scaled by SCALE).
// ---------------------------------------------------------------------------
__global__ __launch_bounds__(256) void k_norm_qkv(
    const float* __restrict__ x, const float* __restrict__ g,
    const float* __restrict__ wqkv, float* __restrict__ qkv) {
  __shared__ float xn[64 * 65];   // stride 65: conflict-free column reads
  __shared__ float nrm[64];
  const int tid = threadIdx.x;
  const int bb = blockIdx.x >> 6;
  const int p0 = (blockIdx.x & 63) << 6;
  const float* xb = x + (size_t)bb * C_IN * NPOS;

  for (int idx = tid; idx < 64 * 64; idx += 256) {
    int c = idx >> 6, p = idx & 63;
    xn[c * 65 + p] = xb[(size_t)c * NPOS + p0 + p];
  }
  __syncthreads();
  if (tid < 64) {
    float s = 0.f;
    #pragma unroll 8
    for (int c = 0; c < 64; ++c) { float v = xn[c * 65 + tid]; s += v * v; }
    nrm[tid] = 8.0f / fmaxf(sqrtf(s), 1e-12f);   // sqrt(64) / max(||x||, eps)
  }
  __syncthreads();
  for (int idx = tid; idx < 64 * 64; idx += 256) {
    int c = idx >> 6, p = idx & 63;
    xn[c * 65 + p] *= nrm[p] * g[c];
  }
  __syncthreads();

  const int wv = tid >> 5, lane = tid & 31, half = lane >> 4, l = lane & 15;
  float* out = qkv + (size_t)bb * NQKV * NPOS;
  for (int mi = 0; mi < 3; ++mi) {
    const int mt = wv * 3 + mi;                  // 24 row-tiles of w_qkv
    const int row0 = mt << 4;
    const float fac = (mt < 8) ? SCALE : 1.0f;   // fold q-scale into Q rows
    v2f a[16];                                   // A frags for all 16 k-steps
    #pragma unroll
    for (int k = 0; k < 16; ++k)
      a[k] = *(const v2f*)(wqkv + (row0 + l) * 64 + (k << 2) + 2 * half);
    for (int nt = 0; nt < 4; ++nt) {
      v8f acc = {};
      #pragma unroll
      for (int k = 0; k < 16; ++k) {
        const int kk = (k << 2) + 2 * half;
        v2f b;
        b.x = xn[kk * 65 + (nt << 4) + l];
        b.y = xn[(kk + 1) * 65 + (nt << 4) + l];
        acc = wmma4(a[k], b, acc);
      }
      #pragma unroll
      for (int r = 0; r < 8; ++r)
        out[(size_t)(row0 + r + 8 * half) * NPOS + p0 + (nt << 4) + l] =
            acc[r] * fac;
    }
  }
}

// ---------------------------------------------------------------------------
// Kernel 2: M[b][h] = K V^T  (32x32, reduce over n=4096).
// One block per (b,h); 8 waves each reduce a 512-wide n-chunk, combine in LDS.
// ---------------------------------------------------------------------------
__global__ __launch_bounds__(256) void k_kv(
    const float* __restrict__ qkv, float* __restrict__ M) {
  __shared__ float red[32 * 32];
  const int tid = threadIdx.x;
  const int bh = blockIdx.x;                     // 0..7
  const int bb = bh >> 2, h = bh & 3;
  const float* Kp = qkv + (size_t)bb * NQKV * NPOS + (size_t)(128 + h * 32) * NPOS;
  const float* Vp = qkv + (size_t)bb * NQKV * NPOS + (size_t)(256 + h * 32) * NPOS;
  const int wv = tid >> 5, lane = tid & 31, half = lane >> 4, l = lane & 15;

  for (int i = tid; i < 1024; i += 256) red[i] = 0.f;
  __syncthreads();

  const int nb0 = wv * 512;
  for (int dt = 0; dt < 2; ++dt)
    for (int pt = 0; pt < 2; ++pt) {
      v8f acc = {};
      const float* Ka = Kp + (size_t)(dt * 16 + l) * NPOS + nb0 + 2 * half;
      const float* Va = Vp + (size_t)(pt * 16 + l) * NPOS + nb0 + 2 * half;
      #pragma unroll 4
      for (int k = 0; k < 128; ++k) {            // 128 * 4 = 512 n per wave
        v2f a = *(const v2f*)(Ka + (k << 2));
        v2f b = *(const v2f*)(Va + (k << 2));
        acc = wmma4(a, b, acc);
      }
      #pragma unroll
      for (int r = 0; r < 8; ++r)
        atomicAdd(&red[(dt * 16 + r + 8 * half) * 32 + pt * 16 + l], acc[r]);
    }
  __syncthreads();
  float* Mg = M + (size_t)bh * 1024;
  for (int i = tid; i < 1024; i += 256) Mg[i] = red[i];
}

// ---------------------------------------------------------------------------
// Kernel 3: hidden = M^T Q (per head), y = w_out*hidden + b_out + x (residual)
// Block = 256 threads, 64 positions of one batch.
// ---------------------------------------------------------------------------
__global__ __launch_bounds__(256) void k_out(
    const float* __restrict__ qkv, const float* __restrict__ M,
    const float* __restrict__ wout, const float* __restrict__ bout,
    const float* __restrict__ x, float* __restrict__ y) {
  __shared__ float Ml[4 * 1024];                 // M for all 4 heads
  __shared__ float Hl[128 * 68];                 // hidden[hd][p], stride 68
  const int tid = threadIdx.x;
  const int bb = blockIdx.x >> 6;
  const int p0 = (blockIdx.x & 63) << 6;

  for (int i = tid; i < 4096; i += 256) Ml[i] = M[(size_t)bb * 4096 + i];
  __syncthreads();

  const int wv = tid >> 5, lane = tid & 31, half = lane >> 4, l = lane & 15;
  const float* Qb = qkv + (size_t)bb * NQKV * NPOS;

  // GEMM1: hidden[h*32+d'][p] = sum_d M[h][d][d'] * Qs[h*32+d][p]
  for (int j = wv * 4; j < wv * 4 + 4; ++j) {    // 32 jobs: head x d'tile x ptile
    const int h = j >> 3, dpt = (j >> 2) & 1, pt = j & 3;
    const float* Mh = Ml + h * 1024;
    const float* Qh = Qb + (size_t)(h * 32) * NPOS + p0;
    v8f acc = {};
    #pragma unroll
    for (int k = 0; k < 8; ++k) {
      const int kk = (k << 2) + 2 * half;
      v2f a, b;
      a.x = Mh[kk * 32 + dpt * 16 + l];          // A[d'][d] = M[d][d'] (transp.)
      a.y = Mh[(kk + 1) * 32 + dpt * 16 + l];
      b.x = Qh[(size_t)kk * NPOS + pt * 16 + l];
      b.y = Qh[(size_t)(kk + 1) * NPOS + pt * 16 + l];
      acc = wmma4(a, b, acc);
    }
    #pragma unroll
    for (int r = 0; r < 8; ++r)
      Hl[(h * 32 + dpt * 16 + r + 8 * half) * 68 + pt * 16 + l] = acc[r];
  }
  __syncthreads();

  // GEMM2: y[o][p] = sum_hd w_out[o][hd] * hidden[hd][p] + b_out[o] + x[o][p]
  for (int j = wv * 2; j < wv * 2 + 2; ++j) {    // 16 jobs: o-tile x p-tile
    const int ot = j >> 2, pt = j & 3;
    v8f acc = {};
    #pragma unroll 4
    for (int k = 0; k < 32; ++k) {               // K = 128
      const int kk = (k << 2) + 2 * half;
      v2f a = *(const v2f*)(wout + (ot * 16 + l) * 128 + kk);
      v2f b;
      b.x = Hl[kk * 68 + pt * 16 + l];
      b.y = Hl[(kk + 1) * 68 + pt * 16 + l];
      acc = wmma4(a, b, acc);
    }
    #pragma unroll
    for (int r = 0; r < 8; ++r) {
      const int o = ot * 16 + r + 8 * half;
      const int p = p0 + pt * 16 + l;
      const size_t oi = (size_t)(bb * 64 + o) * NPOS + p;
      y[oi] = acc[r] + bout[o] + x[oi];
    }
  }
}

// ---------------------------------------------------------------------------
extern "C" void kernel_launch(void* const* d_in, const int* in_sizes, int n_in,
                              void* d_out, int out_size, void* d_ws, size_t ws_size,
                              hipStream_t stream) {
  (void)in_sizes; (void)n_in; (void)out_size; (void)ws_size;
  const float* x    = (const float*)d_in[0];
  const float* g    = (const float*)d_in[1];
  const float* wqkv = (const float*)d_in[2];
  const float* wout = (const float*)d_in[3];
  const float* bout = (const float*)d_in[4];
  float* y   = (float*)d_out;
  float* qkv = (float*)d_ws;                               // 2*384*4096 fp32
  float* M   = qkv + (size_t)NB * NQKV * NPOS;             // 8*32*32 fp32

  k_norm_qkv<<<NB * (NPOS / 64), 256, 0, stream>>>(x, g, wqkv, qkv);
  k_kv<<<NB * NHEAD, 256, 0, stream>>>(qkv, M);
  k_out<<<NB * (NPOS / 64), 256, 0, stream>>>(qkv, M, wout, bout, x, y);
}